// MGGCN_Cell_32272384263006
// MI455X (gfx1250) — compile-verified
//
#include <hip/hip_runtime.h>

// Problem constants (match reference)
#define N_NODES 500
#define BATCH   8
#define FIN0    8
#define T_STEPS 12
#define HW      16          // H1=H2=H3
#define HID     8
#define E_EDGES 8000
#define HG      4000        // N*HID
#define IG      8000        // N*H3
#define BT      96          // BATCH*T_STEPS (batched timesteps)
#define G3W     12000       // 3*HG

typedef __attribute__((ext_vector_type(2))) float v2f;
typedef __attribute__((ext_vector_type(8))) float v8f;

// ---------------- CDNA5 async-to-LDS helpers ----------------

__device__ __forceinline__ void async_copy_b128(unsigned lds_byte_off, const float* gaddr) {
    // per-lane 16B copy global -> LDS, tracked by ASYNCcnt (gfx1250)
    asm volatile("global_load_async_to_lds_b128 %0, %1, off"
                 :: "v"(lds_byte_off), "v"(gaddr) : "memory");
}
__device__ __forceinline__ void wait_asynccnt0() {
    asm volatile("s_wait_asynccnt 0x0" ::: "memory");
}

// ---------------- utility kernels ----------------

__global__ void zero_f32(float* __restrict__ p, int n) {
    int i = blockIdx.x * blockDim.x + threadIdx.x;
    if (i < n) p[i] = 0.0f;
}

__global__ void degree_kernel(const int* __restrict__ src, const int* __restrict__ dst,
                              float* __restrict__ deg_out, float* __restrict__ deg_in) {
    int e = blockIdx.x * blockDim.x + threadIdx.x;
    if (e < E_EDGES) {
        atomicAdd(&deg_out[src[e]], 1.0f);
        atomicAdd(&deg_in[dst[e]], 1.0f);
    }
}

__global__ void norm_kernel(float* __restrict__ n_out, float* __restrict__ n_in) {
    int i = blockIdx.x * blockDim.x + threadIdx.x;
    if (i < N_NODES) {
        n_out[i] = rsqrtf(fmaxf(n_out[i], 1.0f));
        n_in[i]  = rsqrtf(fmaxf(n_in[i],  1.0f));
    }
}

// inputs [N, B, F_IN, T] -> X0 [N, T, B, F_IN]
__global__ void reorder_kernel(const float* __restrict__ inp, float* __restrict__ X0) {
    int i = blockIdx.x * blockDim.x + threadIdx.x;
    const int total = N_NODES * T_STEPS * BATCH * FIN0;
    if (i >= total) return;
    int f = i % FIN0;
    int b = (i / FIN0) % BATCH;
    int t = (i / (FIN0 * BATCH)) % T_STEPS;
    int n = i / (FIN0 * BATCH * T_STEPS);
    X0[i] = inp[((size_t)(n * BATCH + b) * FIN0 + f) * T_STEPS + t];
}

// ---------------- GCN kernels (batched over BT=B*T) ----------------

template<int FIN>
__global__ void gcn_scatter(const float* __restrict__ Xin, const int* __restrict__ src,
                            const int* __restrict__ dst, const float* __restrict__ n_out,
                            float* __restrict__ agg) {
    const int per = BT * FIN;
    int i = blockIdx.x * blockDim.x + threadIdx.x;
    if (i >= E_EDGES * per) return;
    int e = i / per;
    int q = i - e * per;
    int s = src[e];
    atomicAdd(&agg[(size_t)dst[e] * per + q], Xin[(size_t)s * per + q] * n_out[s]);
}

template<int FIN, bool RELU>
__global__ void gcn_dense(const float* __restrict__ agg, const float* __restrict__ n_in,
                          const float* __restrict__ W, const float* __restrict__ bias,
                          float* __restrict__ out) {
    int i = blockIdx.x * blockDim.x + threadIdx.x;
    const int total = N_NODES * BT * HW;
    if (i >= total) return;
    int o  = i % HW;
    int bt = (i / HW) % BT;
    int n  = i / (HW * BT);
    const float* a = agg + (size_t)(n * BT + bt) * FIN;
    float s = 0.0f;
#pragma unroll
    for (int f = 0; f < FIN; ++f) s += a[f] * W[f * HW + o];
    s = s * n_in[n] + bias[o];
    if (RELU) s = fmaxf(s, 0.0f);
    out[(size_t)(n * BT + bt) * HW + o] = s;
}

// g3 [N, T, B, 16] -> xseq [(t*B+b), N*16]
__global__ void build_xseq(const float* __restrict__ g3, float* __restrict__ xseq) {
    int i = blockIdx.x * blockDim.x + threadIdx.x;
    const int total = BT * IG;
    if (i >= total) return;
    int c   = i % HW;
    int n   = (i / HW) % N_NODES;
    int row = i / IG;            // t*B + b
    int t = row / BATCH;
    int b = row - t * BATCH;
    xseq[i] = g3[((size_t)(n * T_STEPS + t) * BATCH + b) * HW + c];
}

// ---------------- Gi GEMM: C[96,Nc] = A[96,K] * W[Nc,K]^T + bias ----------------
// 8 waves/block, each wave a 16-wide column tile, 6 accumulator tiles (all 96 rows).
// A chunks (96 x 32 K-values) double-buffered in LDS via async-to-LDS.
// W chunk (8 K-steps) buffered in registers, NT-hinted; chunks processed in even/odd
// pairs with two *named* register buffers -> no rotation movs, no per-step branches.

#define KB        32                 // K-values per chunk (8 K-steps)
#define LDSSTRIDE (KB + 4)           // 36 floats: 36r mod 64 -> conflict-free ds_load_b64

__device__ __forceinline__ void gi_stage(const float* __restrict__ A, int K, int kc,
                                         unsigned ldsbase, int tid) {
    // 96*32 floats = 768 x b128 / 256 threads = 3 each
#pragma unroll
    for (int i = 0; i < 3; ++i) {
        int e = tid + 256 * i;
        int row = e >> 3, kk4 = e & 7;
        async_copy_b128(ldsbase + (unsigned)(row * LDSSTRIDE + kk4 * 4) * 4u,
                        A + (size_t)row * K + kc + kk4 * 4);
    }
}

__device__ __forceinline__ void gi_loadW(const float* __restrict__ wp, int kc, v2f (&bf)[8]) {
#pragma unroll
    for (int s = 0; s < 8; ++s)
        bf[s] = __builtin_nontemporal_load((const v2f*)(wp + kc + 4 * s));
}

__device__ __forceinline__ void gi_compute(const float* __restrict__ shp, int l15, int half,
                                           const v2f (&bf)[8], v8f (&acc)[6]) {
#pragma unroll
    for (int ks = 0; ks < 8; ++ks)
#pragma unroll
        for (int m = 0; m < 6; ++m) {
            v2f af = *(const v2f*)(shp + (m * 16 + l15) * LDSSTRIDE + ks * 4 + 2 * half);
            acc[m] = __builtin_amdgcn_wmma_f32_16x16x4_f32(
                false, af, false, bf[ks], (short)0, acc[m], false, false);
        }
}

__global__ __launch_bounds__(256) void gemm_gi_wmma(
    const float* __restrict__ A, const float* __restrict__ W,
    const float* __restrict__ bias, float* __restrict__ C,
    int K, int Ncols)
{
    __shared__ float shA[2][BT * LDSSTRIDE];

    const int tid  = threadIdx.x;
    const int lane = tid & 31;
    int wave = blockIdx.x * 8 + (tid >> 5);
    int j0 = wave * 16;
    if (j0 > Ncols - 16) j0 = Ncols - 16;      // clamp (dup work), keep all waves in barriers

    const int half = lane >> 4;
    const int l15  = lane & 15;
    const float* wp = W + (size_t)(j0 + l15) * K + 2 * half;

    unsigned lb0 = (unsigned)(uintptr_t)(&shA[0][0]);
    unsigned lb1 = (unsigned)(uintptr_t)(&shA[1][0]);

    v8f acc[6];
#pragma unroll
    for (int m = 0; m < 6; ++m)
#pragma unroll
        for (int v = 0; v < 8; ++v) acc[m][v] = 0.0f;

    v2f bf0[8], bf1[8];

    const int NCHUNK = K / KB;                 // 250 (even)

    gi_stage(A, K, 0, lb0, tid);
    gi_loadW(wp, 0, bf0);
    wait_asynccnt0();
    __syncthreads();

    for (int c = 0; c < NCHUNK - 2; c += 2) {
        // even chunk c: compute from lds0/bf0, prefetch chunk c+1 into lds1/bf1
        gi_stage(A, K, (c + 1) * KB, lb1, tid);
        gi_loadW(wp, (c + 1) * KB, bf1);
        gi_compute(&shA[0][0], l15, half, bf0, acc);
        wait_asynccnt0();
        __syncthreads();
        // odd chunk c+1: compute from lds1/bf1, prefetch chunk c+2 into lds0/bf0
        gi_stage(A, K, (c + 2) * KB, lb0, tid);
        gi_loadW(wp, (c + 2) * KB, bf0);
        gi_compute(&shA[1][0], l15, half, bf1, acc);
        wait_asynccnt0();
        __syncthreads();
    }
    // chunks NCHUNK-2 (in lds0/bf0) and NCHUNK-1
    gi_stage(A, K, (NCHUNK - 1) * KB, lb1, tid);
    gi_loadW(wp, (NCHUNK - 1) * KB, bf1);
    gi_compute(&shA[0][0], l15, half, bf0, acc);
    wait_asynccnt0();
    __syncthreads();
    gi_compute(&shA[1][0], l15, half, bf1, acc);

    float bv = bias[j0 + l15];
#pragma unroll
    for (int m = 0; m < 6; ++m)
#pragma unroll
        for (int v = 0; v < 8; ++v) {
            int row = m * 16 + v + 8 * half;   // C/D layout: VGPR v -> M=v (+8 hi half)
            C[(size_t)row * Ncols + (j0 + l15)] = acc[m][v] + bv;
        }
}

// ---------------- Gh GEMM: C[16,Nc] = A[16,K] * W[Nc,K]^T + bias ----------------
// Paired-chunk register pipeline (10 K-steps = 40 K-values per chunk, 20 loads in
// flight), two alternating accumulators; W_hh stays RT so the 192MB matrix lives
// in MI455X's 192MB L2 across the 12 recurrent steps.

#define GH_KS 10                               // K-steps per chunk; 4000/(10*4)=100 chunks (even)

__device__ __forceinline__ void gh_load(const float* __restrict__ wp, const float* __restrict__ ap,
                                        int kc, v2f (&bw)[GH_KS], v2f (&ba)[GH_KS]) {
#pragma unroll
    for (int s = 0; s < GH_KS; ++s) {
        bw[s] = *(const v2f*)(wp + kc + 4 * s);
        ba[s] = *(const v2f*)(ap + kc + 4 * s);
    }
}

__device__ __forceinline__ void gh_compute(const v2f (&bw)[GH_KS], const v2f (&ba)[GH_KS],
                                           v8f& acc0, v8f& acc1) {
#pragma unroll
    for (int s = 0; s < GH_KS; ++s) {
        if (s & 1)
            acc1 = __builtin_amdgcn_wmma_f32_16x16x4_f32(
                false, ba[s], false, bw[s], (short)0, acc1, false, false);
        else
            acc0 = __builtin_amdgcn_wmma_f32_16x16x4_f32(
                false, ba[s], false, bw[s], (short)0, acc0, false, false);
    }
}

__global__ __launch_bounds__(256) void gemm_gh_wmma(
    const float* __restrict__ A, const float* __restrict__ W,
    const float* __restrict__ bias, float* __restrict__ C,
    int K, int Ncols)
{
    int wave = (blockIdx.x * blockDim.x + threadIdx.x) >> 5;
    int lane = threadIdx.x & 31;
    int j0 = wave * 16;
    if (j0 >= Ncols) return;                   // wave-uniform exit

    int half = lane >> 4;
    int l15  = lane & 15;
    const float* wp = W + (size_t)(j0 + l15) * K + 2 * half;
    const float* ap = A + (size_t)l15 * K + 2 * half;

    v8f acc0, acc1;
#pragma unroll
    for (int v = 0; v < 8; ++v) { acc0[v] = 0.0f; acc1[v] = 0.0f; }

    v2f bw0[GH_KS], ba0[GH_KS], bw1[GH_KS], ba1[GH_KS];

    const int CK = GH_KS * 4;                  // 40 K-values per chunk
    const int NC = K / CK;                     // 100 (even)

    gh_load(wp, ap, 0, bw0, ba0);
    for (int c = 0; c < NC - 2; c += 2) {
        gh_load(wp, ap, (c + 1) * CK, bw1, ba1);
        gh_compute(bw0, ba0, acc0, acc1);      // chunk c
        gh_load(wp, ap, (c + 2) * CK, bw0, ba0);
        gh_compute(bw1, ba1, acc0, acc1);      // chunk c+1
    }
    gh_load(wp, ap, (NC - 1) * CK, bw1, ba1);
    gh_compute(bw0, ba0, acc0, acc1);          // chunk NC-2
    gh_compute(bw1, ba1, acc0, acc1);          // chunk NC-1

    float bv = bias[j0 + l15];
#pragma unroll
    for (int v = 0; v < 8; ++v) {
        int row = v + 8 * half;
        C[(size_t)row * Ncols + (j0 + l15)] = acc0[v] + acc1[v] + bv;
    }
}

// ---------------- GRU gate fusion ----------------

__global__ void gru_gate(const float* __restrict__ Gi, const float* __restrict__ Gh,
                         float* __restrict__ Hpad, int t)
{
    int i = blockIdx.x * blockDim.x + threadIdx.x;
    if (i >= BATCH * HG) return;
    int b = i / HG;
    int u = i - b * HG;
    const float* gi = Gi + (size_t)(t * BATCH + b) * G3W;
    const float* gh = Gh + (size_t)b * G3W;
    float ir = gi[u], iz = gi[u + HG], inn = gi[u + 2 * HG];
    float hr = gh[u], hz = gh[u + HG], hn  = gh[u + 2 * HG];
    float h  = Hpad[(size_t)b * HG + u];
    float r = 1.0f / (1.0f + __expf(-(ir + hr)));
    float z = 1.0f / (1.0f + __expf(-(iz + hz)));
    float n = tanhf(inn + r * hn);
    Hpad[(size_t)b * HG + u] = (1.0f - z) * n + z * h;
}

__global__ void copy_out(const float* __restrict__ Hpad, float* __restrict__ out) {
    int i = blockIdx.x * blockDim.x + threadIdx.x;
    if (i < BATCH * HG) out[i] = Hpad[i];      // [B, HG] == [B, N, HID] flat
}

// ---------------- launcher ----------------

static inline int nblk(int n, int bs) { return (n + bs - 1) / bs; }

extern "C" void kernel_launch(void* const* d_in, const int* in_sizes, int n_in,
                              void* d_out, int out_size, void* d_ws, size_t ws_size,
                              hipStream_t stream)
{
    const float* inputs = (const float*)d_in[0];
    const int*   src    = (const int*)  d_in[1];
    const int*   dst    = (const int*)  d_in[2];
    const float* W1     = (const float*)d_in[3];
    const float* b1     = (const float*)d_in[4];
    const float* W2     = (const float*)d_in[5];
    const float* b2     = (const float*)d_in[6];
    const float* W3     = (const float*)d_in[7];
    const float* b3     = (const float*)d_in[8];
    const float* W_ih   = (const float*)d_in[9];
    const float* W_hh   = (const float*)d_in[10];
    const float* b_ih   = (const float*)d_in[11];
    const float* b_hh   = (const float*)d_in[12];
    float* out = (float*)d_out;

    // workspace carve-up (floats)
    float* ws    = (float*)d_ws;
    float* n_out = ws;                    // 512
    float* n_in_ = ws + 512;              // 512
    float* X0    = ws + 1024;             // 384000
    float* agg   = X0   + 384000;         // 768000 (max)
    float* h1    = agg  + 768000;
    float* h2    = h1   + 768000;
    float* g3    = h2   + 768000;
    float* xseq  = g3   + 768000;         // 768000
    float* Gi    = xseq + 768000;         // 1152000
    float* Gh    = Gi   + 1152000;        // 192000
    float* Hpad  = Gh   + 192000;         // 64000 (rows 8..15 stay 0)

    const int BS = 256;

    // degree norms
    zero_f32<<<nblk(1024, BS), BS, 0, stream>>>(ws, 1024);
    degree_kernel<<<nblk(E_EDGES, BS), BS, 0, stream>>>(src, dst, n_out, n_in_);
    norm_kernel<<<nblk(N_NODES, BS), BS, 0, stream>>>(n_out, n_in_);

    // input reorder, all timesteps batched
    const int nX0 = N_NODES * BT * FIN0;
    reorder_kernel<<<nblk(nX0, BS), BS, 0, stream>>>(inputs, X0);

    // GCN layer 1: FIN0 -> 16, relu
    zero_f32<<<nblk(nX0, BS), BS, 0, stream>>>(agg, nX0);
    gcn_scatter<FIN0><<<nblk(E_EDGES * BT * FIN0, BS), BS, 0, stream>>>(X0, src, dst, n_out, agg);
    gcn_dense<FIN0, true><<<nblk(N_NODES * BT * HW, BS), BS, 0, stream>>>(agg, n_in_, W1, b1, h1);

    // GCN layer 2: 16 -> 16, relu
    const int n16 = N_NODES * BT * HW;
    zero_f32<<<nblk(n16, BS), BS, 0, stream>>>(agg, n16);
    gcn_scatter<HW><<<nblk(E_EDGES * BT * HW, BS), BS, 0, stream>>>(h1, src, dst, n_out, agg);
    gcn_dense<HW, true><<<nblk(N_NODES * BT * HW, BS), BS, 0, stream>>>(agg, n_in_, W2, b2, h2);

    // GCN layer 3: 16 -> 16, no relu
    zero_f32<<<nblk(n16, BS), BS, 0, stream>>>(agg, n16);
    gcn_scatter<HW><<<nblk(E_EDGES * BT * HW, BS), BS, 0, stream>>>(h2, src, dst, n_out, agg);
    gcn_dense<HW, false><<<nblk(N_NODES * BT * HW, BS), BS, 0, stream>>>(agg, n_in_, W3, b3, g3);

    // sequence rows for the batched input-GEMM
    build_xseq<<<nblk(BT * IG, BS), BS, 0, stream>>>(g3, xseq);

    // Gi = xseq @ W_ih^T + b_ih : 750 column tiles -> 94 blocks x 8 waves (clamped)
    gemm_gi_wmma<<<nblk(750, 8), BS, 0, stream>>>(xseq, W_ih, b_ih, Gi, IG, G3W);

    // GRU recurrence
    zero_f32<<<nblk(16 * HG, BS), BS, 0, stream>>>(Hpad, 16 * HG);
    for (int t = 0; t < T_STEPS; ++t) {
        gemm_gh_wmma<<<nblk(750 * 32, BS), BS, 0, stream>>>(Hpad, W_hh, b_hh, Gh, HG, G3W);
        gru_gate<<<nblk(BATCH * HG, BS), BS, 0, stream>>>(Gi, Gh, Hpad, t);
    }

    copy_out<<<nblk(BATCH * HG, BS), BS, 0, stream>>>(Hpad, out);
}